// ActionRecognition_68375879352919
// MI455X (gfx1250) — compile-verified
//
#include <hip/hip_runtime.h>
#include <hip/hip_bf16.h>

// Problem constants (fixed by the reference)
#define BATCH 256
#define FRAMES 300
#define JC 1600          // feature dim (K of the classifier GEMM)
#define NC 60            // num classes
#define JC4 (JC / 4)     // 400 float4 columns per frame row

typedef __attribute__((ext_vector_type(4))) float v4f;
typedef __attribute__((ext_vector_type(2))) float v2f;
typedef __attribute__((ext_vector_type(8))) float v8f;

// ---------------------------------------------------------------------------
// Kernel 1: masked prefix-mean over frames.  One block per batch sample,
// 400 threads, one float4 column per thread.  Streams x (491 MB) exactly once
// with non-temporal loads so L2 keeps W / pooled hot.
// ---------------------------------------------------------------------------
__global__ void __launch_bounds__(JC4)
ar_pool_kernel(const float* __restrict__ x,
               const int* __restrict__ lengths,
               float* __restrict__ pooled) {
    const int b = blockIdx.x;
    const int t = threadIdx.x;                       // float4 column index, 0..399
    const v4f* xb = (const v4f*)(x + (size_t)b * FRAMES * JC);
    v4f* pout = (v4f*)pooled + (size_t)b * JC4;

    const int len = lengths[b];

    if (len <= 1) {
        // reference edge case: <=1 valid frames -> take frame 0 directly
        pout[t] = __builtin_nontemporal_load(&xb[t]);
        return;
    }

    v4f a0 = (v4f)0.0f, a1 = (v4f)0.0f, a2 = (v4f)0.0f, a3 = (v4f)0.0f;
    int f = 0;
    const int len4 = len & ~3;
    for (; f < len4; f += 4) {
        a0 += __builtin_nontemporal_load(&xb[(size_t)(f + 0) * JC4 + t]);
        a1 += __builtin_nontemporal_load(&xb[(size_t)(f + 1) * JC4 + t]);
        a2 += __builtin_nontemporal_load(&xb[(size_t)(f + 2) * JC4 + t]);
        a3 += __builtin_nontemporal_load(&xb[(size_t)(f + 3) * JC4 + t]);
    }
    for (; f < len; ++f) {
        a0 += __builtin_nontemporal_load(&xb[(size_t)f * JC4 + t]);
    }
    const float inv = 1.0f / (float)len;
    pout[t] = (a0 + a1 + a2 + a3) * inv;
}

// ---------------------------------------------------------------------------
// Kernel 2: out[256,60] = pooled[256,1600] @ W[1600,60] + bias
// via V_WMMA_F32_16X16X4_F32.  One wave per 16x16 output tile.
//   grid.x  = 16  (M tiles, 256/16)
//   block   = (32, 4): threadIdx.y = N tile (N=60 padded to 64)
// A-frag (16x4 f32, 2 VGPR): lane<16 -> row=lane,   K = k+0,k+1
//                            lane>=16 -> row=lane-16, K = k+2,k+3
// B-frag (4x16 f32, 2 VGPR): same halves over K, column = lane&15
// C/D (16x16 f32, 8 VGPR):   VGPR r -> row r + 8*(lane>=16), col = lane&15
// ---------------------------------------------------------------------------
__global__ void __launch_bounds__(128)
ar_gemm_kernel(const float* __restrict__ P,
               const float* __restrict__ W,
               const float* __restrict__ bias,
               float* __restrict__ out) {
    const int lane  = threadIdx.x;        // 0..31
    const int half  = lane >> 4;          // 0 or 1 (selects K pair)
    const int l16   = lane & 15;
    const int m0    = blockIdx.x * 16;    // output row tile
    const int n0    = threadIdx.y * 16;   // output col tile

    const int arow  = m0 + l16;
    const int bcol  = n0 + l16;
    const bool bvalid = (bcol < NC);
    const int bcolL = bvalid ? bcol : (NC - 1);   // clamp load address only;
                                                  // garbage cols never stored

    const float* __restrict__ aptr = P + (size_t)arow * JC + 2 * half;
    const float* __restrict__ wptr = W + (size_t)(2 * half) * NC + bcolL;

    v8f acc = (v8f)0.0f;

    for (int k = 0; k < JC; k += 4) {
        v2f afrag = *(const v2f*)(aptr + k);
        v2f bfrag;
        bfrag.x = wptr[(size_t)k * NC];
        bfrag.y = wptr[(size_t)k * NC + NC];
        acc = __builtin_amdgcn_wmma_f32_16x16x4_f32(
                  /*neg_a=*/false, afrag,
                  /*neg_b=*/false, bfrag,
                  /*c_mod=*/(short)0, acc,
                  /*reuse_a=*/false, /*reuse_b=*/false);
    }

    if (bvalid) {
        const float bv = bias[bcol];
#pragma unroll
        for (int r = 0; r < 8; ++r) {
            out[(size_t)(m0 + r + 8 * half) * NC + bcol] = acc[r] + bv;
        }
    }
}

// ---------------------------------------------------------------------------
// Launch: inputs in reference order: x[f32], lengths[i32], W[f32], b[f32]
// ---------------------------------------------------------------------------
extern "C" void kernel_launch(void* const* d_in, const int* in_sizes, int n_in,
                              void* d_out, int out_size, void* d_ws, size_t ws_size,
                              hipStream_t stream) {
    const float* x       = (const float*)d_in[0];
    const int*   lengths = (const int*)d_in[1];
    const float* W       = (const float*)d_in[2];
    const float* bias    = (const float*)d_in[3];
    float*       out     = (float*)d_out;

    float* pooled = (float*)d_ws;   // BATCH*JC floats = 1.64 MB scratch

    ar_pool_kernel<<<dim3(BATCH), dim3(JC4), 0, stream>>>(x, lengths, pooled);
    ar_gemm_kernel<<<dim3(BATCH / 16), dim3(32, (NC + 15) / 16), 0, stream>>>(
        pooled, W, bias, out);
}